// HausdorffDistance_1116691497266
// MI455X (gfx1250) — compile-verified
//
#include <hip/hip_runtime.h>
#include <hip/hip_bf16.h>

typedef __attribute__((ext_vector_type(2))) float v2f;
typedef __attribute__((ext_vector_type(8))) float v8f;

#define BATCH  16
#define HH     96
#define WW     96
#define NPTS   (HH * WW)      /* 9216 */
#define NTILES (NPTS / 16)    /* 576  */
#define XTILES (WW / 16)      /* 6    */
#define TAU    0.5f
#define BIGF   1e30f

// d_ws layout (32-bit words):
//  [0..15]   d_ab^2 per image (float bits, max-accumulated)
//  [16..31]  d_ba^2
//  [32..47]  diamA^2
//  [48..63]  diamB^2
//  [64]      anyA bit-word (bit k = image k nonempty)
//  [65]      anyB bit-word
//  [66 .. 66+NPTS)          packedA masks (bit k = pred[k][n] >= TAU)
//  [66+NPTS .. 66+2*NPTS)   packedB masks

__global__ __launch_bounds__(128) void hd_init(unsigned* ws) {
    int t = threadIdx.x;
    if (t < 66) ws[t] = 0u;
}

__global__ __launch_bounds__(256) void hd_pack(const float* __restrict__ pred,
                                               const float* __restrict__ targ,
                                               unsigned* __restrict__ ws) {
    int n = blockIdx.x * blockDim.x + threadIdx.x;
    if (n >= NPTS) return;
    unsigned wa = 0u, wb = 0u;
#pragma unroll
    for (int k = 0; k < BATCH; ++k) {
        wa |= (pred[k * NPTS + n] >= TAU ? 1u : 0u) << k;
        wb |= (targ[k * NPTS + n] >= TAU ? 1u : 0u) << k;
    }
    ws[66 + n]        = wa;
    ws[66 + NPTS + n] = wb;
    atomicOr(&ws[64], wa);
    atomicOr(&ws[65], wb);
}

// One block per 16-row tile; wave w (of 16) handles image w.
// Column tiles iterate as (row jy, x-tile jx) so coordinates update
// incrementally. Each tile: one V_WMMA_F32_16X16X4_F32 gives the exact
// 16x16 squared-distance tile; compare-select masked reductions accumulate.
__global__ __launch_bounds__(512) void hd_main(const unsigned* __restrict__ packedA,
                                               const unsigned* __restrict__ packedB,
                                               unsigned* __restrict__ gacc) {
    const int lane = threadIdx.x & 31;
    const int img  = threadIdx.x >> 5;   // 0..15, image index
    const int it   = blockIdx.x;         // row tile 0..575
    const int half = lane >> 4;          // C rows: half0 -> M=r, half1 -> M=8+r
    const int l16  = lane & 15;

    // A operand: row p = it*16 + l16, A[p] = (y, x, |p|^2, 1)
    // ISA layout: vgpr0 = K0 (lanes 0-15) / K2 (lanes 16-31); vgpr1 = K1 / K3
    const int   p   = it * 16 + l16;
    const float yA  = (float)(p / WW);
    const float xA  = (float)(p % WW);
    const float sqA = yA * yA + xA * xA;
    v2f Aop;
    Aop.x = half ? sqA  : yA;
    Aop.y = half ? 1.0f : xA;

    float accMinA[8], accMinB[8], accMaxA[8], accMaxB[8];
#pragma unroll
    for (int r = 0; r < 8; ++r) {
        accMinA[r] = BIGF; accMinB[r] = BIGF;
        accMaxA[r] = 0.0f; accMaxB[r] = 0.0f;
    }

    const unsigned* __restrict__ mAp = packedA + l16;
    const unsigned* __restrict__ mBp = packedB + l16;

    float yB = 0.0f;
    for (int jy = 0; jy < HH; ++jy) {
        const float n2yB  = -2.0f * yB;
        const float yB2   = yB * yB;
        float       xB    = (float)l16;
#pragma unroll
        for (int jx = 0; jx < XTILES; ++jx) {
            const float sqB = yB2 + xB * xB;
            v2f Bop;
            Bop.x = half ? 1.0f : n2yB;
            Bop.y = half ? sqB  : -2.0f * xB;

            v8f c = {};
            c = __builtin_amdgcn_wmma_f32_16x16x4_f32(
                    /*neg_a=*/false, Aop, /*neg_b=*/false, Bop,
                    /*c_mod=*/(short)0, c, /*reuse_a=*/false, /*reuse_b=*/false);

            // column masks for this image (column within tile == l16)
            const int idx = (jy * XTILES + jx) * 16;
            const bool bA = ((mAp[idx] >> img) & 1u) != 0u;
            const bool bB = ((mBp[idx] >> img) & 1u) != 0u;
#pragma unroll
            for (int r = 0; r < 8; ++r) {
                const float d2 = c[r];   // row = it*16 + r + 8*half
                accMinA[r] = (bA && d2 < accMinA[r]) ? d2 : accMinA[r];
                accMinB[r] = (bB && d2 < accMinB[r]) ? d2 : accMinB[r];
                accMaxA[r] = (bA && d2 > accMaxA[r]) ? d2 : accMaxA[r];
                accMaxB[r] = (bB && d2 > accMaxB[r]) ? d2 : accMaxB[r];
            }
            xB += 16.0f;
        }
        yB += 1.0f;
    }

    // Reduce across the 16 lanes of each half (columns) -> per-row totals.
#pragma unroll
    for (int r = 0; r < 8; ++r) {
        for (int s = 1; s < 16; s <<= 1) {
            accMinA[r] = fminf(accMinA[r], __shfl_xor(accMinA[r], s, 32));
            accMinB[r] = fminf(accMinB[r], __shfl_xor(accMinB[r], s, 32));
            accMaxA[r] = fmaxf(accMaxA[r], __shfl_xor(accMaxA[r], s, 32));
            accMaxB[r] = fmaxf(accMaxB[r], __shfl_xor(accMaxB[r], s, 32));
        }
    }

    // Apply row masks, max over this tile's rows.
    float pAB = 0.0f, pBA = 0.0f, pDA = 0.0f, pDB = 0.0f;
#pragma unroll
    for (int r = 0; r < 8; ++r) {
        const int row = it * 16 + r + 8 * half;
        const bool rA = ((packedA[row] >> img) & 1u) != 0u;
        const bool rB = ((packedB[row] >> img) & 1u) != 0u;
        pAB = fmaxf(pAB, (rA ? accMinB[r] : 0.0f));  // sup_{p in A} min_{q in B}
        pBA = fmaxf(pBA, (rB ? accMinA[r] : 0.0f));  // sup_{p in B} min_{q in A}
        pDA = fmaxf(pDA, (rA ? accMaxA[r] : 0.0f));  // diam(A)^2 partial
        pDB = fmaxf(pDB, (rB ? accMaxB[r] : 0.0f));  // diam(B)^2 partial
    }
    // combine the two halves (rows 0-7 vs 8-15)
    pAB = fmaxf(pAB, __shfl_xor(pAB, 16, 32));
    pBA = fmaxf(pBA, __shfl_xor(pBA, 16, 32));
    pDA = fmaxf(pDA, __shfl_xor(pDA, 16, 32));
    pDB = fmaxf(pDB, __shfl_xor(pDB, 16, 32));

    if (lane == 0) {
        // all values >= 0 -> uint compare == float compare
        atomicMax(&gacc[img],      __float_as_uint(pAB));
        atomicMax(&gacc[16 + img], __float_as_uint(pBA));
        atomicMax(&gacc[32 + img], __float_as_uint(pDA));
        atomicMax(&gacc[48 + img], __float_as_uint(pDB));
    }
}

__global__ __launch_bounds__(32) void hd_finalize(const unsigned* __restrict__ gacc,
                                                  float* __restrict__ out) {
    const int lane = threadIdx.x;
    float hd = 0.0f;
    if (lane < 16) {
        const unsigned anyA = (gacc[64] >> lane) & 1u;
        const unsigned anyB = (gacc[65] >> lane) & 1u;
        const float dab2 = __uint_as_float(gacc[lane]);
        const float dba2 = __uint_as_float(gacc[16 + lane]);
        const float da2  = __uint_as_float(gacc[32 + lane]);
        const float db2  = __uint_as_float(gacc[48 + lane]);
        float h2;
        if (anyA && anyB)   h2 = fmaxf(dab2, dba2);
        else if (anyA)      h2 = da2;
        else if (anyB)      h2 = db2;
        else                h2 = 0.0f;
        hd = sqrtf(h2);
    }
    for (int s = 1; s < 16; s <<= 1) hd += __shfl_xor(hd, s, 32);
    if (lane == 0) out[0] = hd * (1.0f / (float)BATCH);
}

extern "C" void kernel_launch(void* const* d_in, const int* in_sizes, int n_in,
                              void* d_out, int out_size, void* d_ws, size_t ws_size,
                              hipStream_t stream) {
    const float* pred = (const float*)d_in[0];
    const float* targ = (const float*)d_in[1];
    unsigned* ws  = (unsigned*)d_ws;
    float*    out = (float*)d_out;
    (void)in_sizes; (void)n_in; (void)out_size; (void)ws_size;

    hd_init<<<1, 128, 0, stream>>>(ws);
    hd_pack<<<(NPTS + 255) / 256, 256, 0, stream>>>(pred, targ, ws);
    hd_main<<<NTILES, 512, 0, stream>>>(ws + 66, ws + 66 + NPTS, ws);
    hd_finalize<<<1, 32, 0, stream>>>(ws, out);
}